// QuantumTokenEncoder_39281770889751
// MI455X (gfx1250) — compile-verified
//
#include <hip/hip_runtime.h>
#include <hip/hip_bf16.h>

// ---------------------------------------------------------------------------
// QuantumTokenEncoder for MI455X (gfx1250).
//
// Math: per token t with angles x[0..3]:
//   s = kron_{w}( (cos(x_w/2), sin(x_w/2)) )           // 16 real amplitudes
//   y = U s            // U = fixed 16x16 orthogonal matrix from `weights`
//   e_w = sum_b sign_w(b) * y_b^2,  sign_w(b) = 1-2*bit_{3-w}(b)
//
// U is built once per launch by a 16-thread prep kernel (columns = circuit
// applied to basis vectors).  Main kernel: one wave32 handles 16 tokens per
// iteration; the 16x16 matvec for those 16 tokens is D = U(16x16) @ S(16x16)
// done as 4 chained V_WMMA_F32_16X16X4_F32.  Squaring + sign contraction is
// 8 VALU muls + signed sums per half-wave + four ds_swizzle SWAPX16.
//
// Traffic: 16 MB total -> ~0.7us floor @ 23.3 TB/s; compute is far under it.
// ---------------------------------------------------------------------------

typedef float v2f __attribute__((ext_vector_type(2)));
typedef float v8f __attribute__((ext_vector_type(8)));

// Cross half-wave swap (lane <-> lane^16) via LDS swizzle unit (no LDS memory
// touched).  Group-of-32 encoding: offset[14:10]=xor, [9:5]=or, [4:0]=and.
__device__ __forceinline__ float swap16(float x) {
    int xi = __builtin_bit_cast(int, x);
    int yi = __builtin_amdgcn_ds_swizzle(xi, (0x10 << 10) | 0x1F);  // SWAPX16
    return __builtin_bit_cast(float, yi);
}

// Build U (16x16, row-major) in d_ws.  Thread j simulates the weight circuit
// on basis state e_j; column j of U is the result.  Wire w <-> bit (3-w).
__global__ void build_u_kernel(const float* __restrict__ w, float* __restrict__ U) {
    int j = threadIdx.x;
    if (j >= 16) return;
    float st[16];
#pragma unroll
    for (int i = 0; i < 16; ++i) st[i] = (i == j) ? 1.0f : 0.0f;
#pragma unroll
    for (int l = 0; l < 2; ++l) {
        // RY(weights[l][wi]) on each wire
#pragma unroll
        for (int wi = 0; wi < 4; ++wi) {
            float sn, cs;
            __sincosf(w[l * 4 + wi] * 0.5f, &sn, &cs);
            int p = 3 - wi;
#pragma unroll
            for (int b = 0; b < 16; ++b) {
                if (((b >> p) & 1) == 0) {
                    int b1 = b | (1 << p);
                    float s0 = st[b], s1 = st[b1];
                    st[b]  = cs * s0 - sn * s1;
                    st[b1] = sn * s0 + cs * s1;
                }
            }
        }
        // CNOT chain: control wi, target wi+1  (swap target bit where control=1)
#pragma unroll
        for (int wi = 0; wi < 3; ++wi) {
            int pc = 3 - wi, pt = 2 - wi;
#pragma unroll
            for (int b = 0; b < 16; ++b) {
                if (((b >> pc) & 1) == 1 && ((b >> pt) & 1) == 0) {
                    int b2 = b | (1 << pt);
                    float tmp = st[b]; st[b] = st[b2]; st[b2] = tmp;
                }
            }
        }
    }
#pragma unroll
    for (int i = 0; i < 16; ++i) U[i * 16 + j] = st[i];
}

__global__ void qenc_kernel(const float* __restrict__ in,
                            const float* __restrict__ U,
                            float* __restrict__ out,
                            int nTok, int nGroups) {
    const int lane = threadIdx.x & 31;
    const int m    = lane & 15;   // row (A) / token-in-group (B,D)
    const int h    = lane >> 4;   // half-wave: selects K sub-rows 2h,2h+1

    // A fragments: a[k] covers K columns 4k..4k+3 of U.
    // f32 16x4 A layout: lanes0-15 V0=K0,V1=K1; lanes16-31 V0=K2,V1=K3.
    v2f a[4];
#pragma unroll
    for (int k = 0; k < 4; ++k) {
        a[k].x = U[m * 16 + 4 * k + 2 * h];
        a[k].y = U[m * 16 + 4 * k + 2 * h + 1];
    }

    const int wave   = (blockIdx.x * blockDim.x + threadIdx.x) >> 5;
    const int nWaves = (gridDim.x * blockDim.x) >> 5;

    for (int g = wave; g < nGroups; g += nWaves) {
        const int t = g * 16 + m;
        float4 x = make_float4(0.f, 0.f, 0.f, 0.f);
        if (t < nTok) x = reinterpret_cast<const float4*>(in)[t];

        float c0, s0, c1, s1, c2, s2, c3, s3;
        __sincosf(x.x * 0.5f, &s0, &c0);
        __sincosf(x.y * 0.5f, &s1, &c1);
        __sincosf(x.z * 0.5f, &s2, &c2);
        __sincosf(x.w * 0.5f, &s3, &c3);

        // Amplitude s_b, b = (k1 k0 h j) bits = wire0..wire3 selections.
        // This lane supplies B rows K=4k+2h (j=0) and 4k+2h+1 (j=1).
        const float g2   = h ? s2 : c2;
        const float pre0 = g2 * c3;
        const float pre1 = g2 * s3;
        const float mk[4] = {c0 * c1, c0 * s1, s0 * c1, s0 * s1};

        v8f acc = {0.f, 0.f, 0.f, 0.f, 0.f, 0.f, 0.f, 0.f};
#pragma unroll
        for (int k = 0; k < 4; ++k) {
            v2f b;
            b.x = mk[k] * pre0;
            b.y = mk[k] * pre1;
            // D(16x16) += U[:,4k:4k+4] @ S[4k:4k+4, :]
            acc = __builtin_amdgcn_wmma_f32_16x16x4_f32(
                /*neg_a=*/false, a[k], /*neg_b=*/false, b,
                /*c_mod=*/(short)0, acc, /*reuse_a=*/false, /*reuse_b=*/false);
        }

        // probs for this lane's 8 rows: b = r + 8h
        float q[8];
#pragma unroll
        for (int r = 0; r < 8; ++r) q[r] = acc[r] * acc[r];

        // Signed partial sums over local rows; sign_w(b) = 1-2*bit_{3-w}(b)
        float T   = ((q[0] + q[1]) + (q[2] + q[3])) + ((q[4] + q[5]) + (q[6] + q[7]));
        float e0p = h ? -T : T;                                        // bit3 = h
        float e1p = (q[0] + q[1] + q[2] + q[3]) - (q[4] + q[5] + q[6] + q[7]); // bit2
        float e2p = (q[0] + q[1] + q[4] + q[5]) - (q[2] + q[3] + q[6] + q[7]); // bit1
        float e3p = (q[0] + q[2] + q[4] + q[6]) - (q[1] + q[3] + q[5] + q[7]); // bit0

        // Add the other half-wave's partial (rows 8-15 vs 0-7).
        float e0 = e0p + swap16(e0p);
        float e1 = e1p + swap16(e1p);
        float e2 = e2p + swap16(e2p);
        float e3 = e3p + swap16(e3p);

        if (h == 0 && t < nTok) {
            float4 o;
            o.x = e0; o.y = e1; o.z = e2; o.w = e3;
            reinterpret_cast<float4*>(out)[t] = o;   // coalesced 128-bit store
        }
    }
}

extern "C" void kernel_launch(void* const* d_in, const int* in_sizes, int n_in,
                              void* d_out, int out_size, void* d_ws, size_t ws_size,
                              hipStream_t stream) {
    const float* in = (const float*)d_in[0];   // projected_tokens (128,4096,4) f32
    const float* w  = (const float*)d_in[1];   // weights (2,4) f32
    float* out = (float*)d_out;                // (128,4096,4) f32
    float* U   = (float*)d_ws;                 // 16x16 f32 scratch (1 KB)

    const int nTok    = in_sizes[0] / 4;       // 524288
    const int nGroups = (nTok + 15) / 16;      // 32768 wave-iterations

    hipLaunchKernelGGL(build_u_kernel, dim3(1), dim3(32), 0, stream, w, U);

    // 256 threads = 8 wave32 per block; grid-stride over token groups.
    const int blocks = 2048;                   // 16384 waves -> 2 groups/wave
    hipLaunchKernelGGL(qenc_kernel, dim3(blocks), dim3(256), 0, stream,
                       in, U, out, nTok, nGroups);
}